// yoloLoss_31318901522732
// MI455X (gfx1250) — compile-verified
//
#include <hip/hip_runtime.h>
#include <cstdint>

#define TPB 256
#define CELLS_PER_BLOCK 256
#define CELL_F 30   // floats per cell (B*5 + C = 30)

// ---------------- CDNA5 async global->LDS copies (ASYNCcnt path) ----------------

typedef int v4i_t __attribute__((vector_size(4 * sizeof(int))));
typedef int v2i_t __attribute__((vector_size(2 * sizeof(int))));

__device__ __forceinline__ void async_copy_b128(const float* g, float* l) {
#if __has_builtin(__builtin_amdgcn_global_load_async_to_lds_b128)
  __builtin_amdgcn_global_load_async_to_lds_b128(
      (__attribute__((address_space(1))) v4i_t*)(float*)g,
      (__attribute__((address_space(3))) v4i_t*)l, 0, 0);
#else
  asm volatile("global_load_async_to_lds_b128 %0, %1, off"
               :: "v"((unsigned)(unsigned long long)
                        (__attribute__((address_space(3))) void*)l),
                  "v"((unsigned long long)g)
               : "memory");
#endif
}

__device__ __forceinline__ void async_copy_b64(const float* g, float* l) {
#if __has_builtin(__builtin_amdgcn_global_load_async_to_lds_b64)
  __builtin_amdgcn_global_load_async_to_lds_b64(
      (__attribute__((address_space(1))) v2i_t*)(float*)g,
      (__attribute__((address_space(3))) v2i_t*)l, 0, 0);
#else
  asm volatile("global_load_async_to_lds_b64 %0, %1, off"
               :: "v"((unsigned)(unsigned long long)
                        (__attribute__((address_space(3))) void*)l),
                  "v"((unsigned long long)g)
               : "memory");
#endif
}

__device__ __forceinline__ void wait_async0() {
#if __has_builtin(__builtin_amdgcn_s_wait_asynccnt)
  __builtin_amdgcn_s_wait_asynccnt(0);
#else
  asm volatile("s_wait_asynccnt 0" ::: "memory");
#endif
}

// ---------------- per-cell YOLO loss math ----------------

__device__ __forceinline__ float clip01(float v) {
  return fminf(fmaxf(v, 0.0f), 1.0f);
}

__device__ __forceinline__ float iou_one(const float* pb, const float* tb,
                                         float gx, float gy) {
  const float STEP = 1.0f / 7.0f;
  float x1 = clip01((pb[0] + gx) * STEP - pb[2] * 0.5f);
  float y1 = clip01((pb[1] + gy) * STEP - pb[3] * 0.5f);
  float w1 = clip01(pb[2]);
  float h1 = clip01(pb[3]);
  float x2 = clip01((tb[0] + gx) * STEP - tb[2] * 0.5f);
  float y2 = clip01((tb[1] + gy) * STEP - tb[3] * 0.5f);
  float w2 = clip01(tb[2]);
  float h2 = clip01(tb[3]);
  float iw = fmaxf(w1 + w2 - (fmaxf(x1 + w1, x2 + w2) - fminf(x1, x2)), 0.0f);
  float ih = fmaxf(h1 + h2 - (fmaxf(y1 + h1, y2 + h2) - fminf(y1, y2)), 0.0f);
  float inter = iw * ih;
  float uni = w1 * h1 + w2 * h2 - inter + 1e-6f;
  return inter / uni;
}

__device__ __forceinline__ float cell_loss(const float* p, const float* t,
                                           float gx, float gy) {
  float iou0 = iou_one(p, t, gx, gy);
  float iou1 = iou_one(p + 5, t + 5, gx, gy);
  int best = (iou1 > iou0) ? 1 : 0;   // argmax, tie -> box 0
  float loss = 0.0f;
  bool sig = false;
#pragma unroll
  for (int b = 0; b < 2; ++b) {
    const float* pb = p + b * 5;
    const float* tb = t + b * 5;
    bool o0 = tb[4] > 0.0f;
    sig = sig || o0;
    bool objb = o0 && (b == best);
    float dc = pb[4] - tb[4];
    dc *= dc;
    loss += objb ? dc : 0.5f * dc;    // obj_loss + L_NOOBJ * noobj_loss
    if (objb) {
      float dx = pb[0] - tb[0];
      float dy = pb[1] - tb[1];
      loss += 5.0f * (dx * dx + dy * dy);             // L_COORD * xy
      float pw = fmaxf(pb[2], 0.0f);
      float ph = fmaxf(pb[3], 0.0f);
      float dw = sqrtf(tb[2] + 1e-6f) - sqrtf(pw + 1e-6f);
      float dh = sqrtf(tb[3] + 1e-6f) - sqrtf(ph + 1e-6f);
      loss += 5.0f * (dw * dw + dh * dh);             // L_COORD * wh
    }
  }
  if (sig) {
    float cl = 0.0f;
#pragma unroll
    for (int k = 10; k < 30; ++k) {
      float d = p[k] - t[k];
      cl = fmaf(d, d, cl);
    }
    loss += cl;
  }
  return loss;
}

// ---------------- kernel ----------------

__global__ __launch_bounds__(TPB) void yolo_loss_kernel(
    const float* __restrict__ pred, const float* __restrict__ targ,
    float* __restrict__ out, int numCells, float invN) {
  // 256 cells * 30 floats * 4B = 30720 B per array; + 1KB reduction = 62.5 KB LDS
  __shared__ float sP[CELLS_PER_BLOCK * CELL_F];
  __shared__ float sT[CELLS_PER_BLOCK * CELL_F];
  __shared__ float sRed[TPB];

  const int tid = threadIdx.x;
  const long long cell0 = (long long)blockIdx.x * CELLS_PER_BLOCK;
  float loss = 0.0f;

  if (cell0 + CELLS_PER_BLOCK <= (long long)numCells) {
    // Fast path: stage this wave's 32 contiguous cells (3840 B per array)
    // into LDS with async copies, then each lane reads its own cell.
    const int lane = tid & 31;
    const int wv = tid >> 5;
    const size_t base = (size_t)cell0 * CELL_F + (size_t)wv * (32 * CELL_F);
    const float* gp = pred + base;
    const float* gt = targ + base;
    float* lp = sP + wv * (32 * CELL_F);
    float* lt = sT + wv * (32 * CELL_F);
#pragma unroll
    for (int k = 0; k < 7; ++k) {            // 7 * 32 lanes * 16B = 3584 B
      const int off = lane * 4 + k * 128;    // in floats
      async_copy_b128(gp + off, lp + off);
      async_copy_b128(gt + off, lt + off);
    }
    {                                        // tail: 32 lanes * 8B = 256 B
      const int off = 896 + lane * 2;
      async_copy_b64(gp + off, lp + off);
      async_copy_b64(gt + off, lt + off);
    }
    wait_async0();   // per-wave: only this wave reads this region, no barrier

    const int rem = (int)((cell0 + tid) % 49);
    loss = cell_loss(sP + tid * CELL_F, sT + tid * CELL_F,
                     (float)(rem % 7), (float)(rem / 7));
  } else if (cell0 + tid < (long long)numCells) {
    // Tail block: direct global reads.
    const size_t base = (size_t)(cell0 + tid) * CELL_F;
    const int rem = (int)((cell0 + tid) % 49);
    loss = cell_loss(pred + base, targ + base,
                     (float)(rem % 7), (float)(rem / 7));
  }

  // Block reduction, then one float atomic per block.
  sRed[tid] = loss;
  __syncthreads();
#pragma unroll
  for (int s = TPB / 2; s > 0; s >>= 1) {
    if (tid < s) sRed[tid] += sRed[tid + s];
    __syncthreads();
  }
  if (tid == 0) atomicAdd(out, sRed[0] * invN);
}

// ---------------- launch ----------------

extern "C" void kernel_launch(void* const* d_in, const int* in_sizes, int n_in,
                              void* d_out, int out_size, void* d_ws, size_t ws_size,
                              hipStream_t stream) {
  (void)n_in; (void)out_size; (void)d_ws; (void)ws_size;
  const float* pred = (const float*)d_in[0];
  const float* targ = (const float*)d_in[1];
  float* out = (float*)d_out;

  const int total = in_sizes[0];             // N * 7 * 7 * 30
  const int N = total / (7 * 7 * CELL_F);
  const int numCells = N * 49;
  const int blocks = (numCells + CELLS_PER_BLOCK - 1) / CELLS_PER_BLOCK;

  (void)hipMemsetAsync(out, 0, sizeof(float), stream);
  yolo_loss_kernel<<<blocks, TPB, 0, stream>>>(pred, targ, out, numCells,
                                               1.0f / (float)N);
}